// Atten_tit_1563368096548
// MI455X (gfx1250) — compile-verified
//
#include <hip/hip_runtime.h>

typedef float v2f __attribute__((ext_vector_type(2)));
typedef float v8f __attribute__((ext_vector_type(8)));

#define Bb 16
#define Tt 256
#define Kk 256
#define Hh 512
#define Aa 49
#define AP 64   // padded attention dim (49 -> 64)

// Hardware tanh: V_TANH_F32 is a CDNA5 TRANS32 op. Prefer the builtin (the
// compiler then knows the trans hazard rules); fall back to inline asm with a
// v_nop to satisfy the documented 1-independent-op TRANS->use requirement.
#if defined(__has_builtin) && __has_builtin(__builtin_amdgcn_tanhf)
static __device__ __forceinline__ float fast_tanh(float x) {
    return __builtin_amdgcn_tanhf(x);
}
#else
static __device__ __forceinline__ float fast_tanh(float x) {
    float r;
    asm volatile("v_tanh_f32 %0, %1\n\tv_nop" : "=v"(r) : "v"(x));
    return r;
}
#endif

// ---------------------------------------------------------------------------
// Kernel 1: Out(M x AP) = X(M x Hh) * W(Aa x Hh)^T   with M = Bb*256 = 4096.
// One wave per 16x16 output tile, f32 WMMA, K-loop of Hh/4 = 128 wmma ops.
// Padding lanes (n = 49..63) just re-read row 48 and produce junk columns —
// those columns are never consumed downstream, so no masking is needed and
// all loads stay on the global (LOADcnt-only) path.
// ---------------------------------------------------------------------------
__global__ __launch_bounds__(256) void proj_wmma(const float* __restrict__ X,
                                                 const float* __restrict__ W,
                                                 float* __restrict__ Out) {
    const int tid  = threadIdx.x;
    const int wave = tid >> 5;
    const int lane = tid & 31;
    const int g    = blockIdx.x * 8 + wave;   // 1024 tiles total
    const int m0   = (g >> 2) << 4;           // row tile base
    const int n0   = (g & 3) << 4;            // col tile base (AP/16 = 4)
    const int lr   = lane & 15;
    const int lh   = lane >> 4;               // 0 or 1 (k-pair select)

    const int n  = n0 + lr;
    const int nc = (n < Aa) ? n : (Aa - 1);   // clamp: junk cols never read
    const float* Xr = X + (size_t)(m0 + lr) * Hh;
    const float* Wr = W + (size_t)nc * Hh;

    v8f c = {0.f, 0.f, 0.f, 0.f, 0.f, 0.f, 0.f, 0.f};
    for (int kb = 0; kb < Hh; kb += 4) {
        const int k = kb + (lh << 1);
        v2f a  = *(const v2f*)(Xr + k);   // A: 16x4 f32 fragment (b64 load)
        v2f bf = *(const v2f*)(Wr + k);   // B: 4x16 f32 fragment (b64 load)
        c = __builtin_amdgcn_wmma_f32_16x16x4_f32(false, a, false, bf,
                                                  (short)0, c, false, false);
    }
    const int mrow = m0 + (lh << 3);
#pragma unroll
    for (int v = 0; v < 8; ++v)
        Out[(size_t)(mrow + v) * AP + n0 + lr] = c[v];
}

// ---------------------------------------------------------------------------
// Kernel 2: per (b, 16-row t-tile):
//   z[tt][k] = sum_a tanh(cv[b,k,a] + cg[b,t,a]) * Wh[a]   (V_TANH_F32 trans)
//   alpha    = softmax_k(z)                                 (wave32 shuffles)
//   c_hat    = alpha(16x256) * des_b(256x512)               (f32 WMMA)
// ---------------------------------------------------------------------------
__global__ __launch_bounds__(256) void attn_wmma(const float* __restrict__ cv,
                                                 const float* __restrict__ cg,
                                                 const float* __restrict__ Wh,
                                                 const float* __restrict__ des,
                                                 float* __restrict__ chat,
                                                 float* __restrict__ alpha) {
    __shared__ float zbuf[16][Kk];   // 16 KB: z then alpha tile
    __shared__ float cgl[16][AP];    //  4 KB: staged cg tile
    __shared__ float whl[AP];

    const int tid = threadIdx.x;
    const int b   = blockIdx.x >> 4;
    const int t0  = (blockIdx.x & 15) << 4;

    // --- stage cg tile + Wh into LDS ---
    for (int idx = tid; idx < 16 * Aa; idx += 256) {
        const int tt = idx / Aa, a = idx % Aa;
        cgl[tt][a] = cg[(size_t)(b * Tt + t0 + tt) * AP + a];
    }
    if (tid < AP) whl[tid] = (tid < Aa) ? Wh[tid] : 0.f;
    __syncthreads();

    // --- z tile: each thread owns row tt = tid/16, 16 k columns ---
    {
        const int tt = tid >> 4;
        const int kc = tid & 15;
        for (int i = 0; i < 16; ++i) {
            const int k = kc + (i << 4);
            const float4* cvp = (const float4*)(cv + (size_t)(b * Kk + k) * AP);
            float z = 0.f;
#pragma unroll 4
            for (int q = 0; q < 12; ++q) {          // a = 0..47
                const float4 cvv = cvp[q];
                const float* cgr = &cgl[tt][q * 4];
                const float* whr = &whl[q * 4];
                z += fast_tanh(cvv.x + cgr[0]) * whr[0];
                z += fast_tanh(cvv.y + cgr[1]) * whr[1];
                z += fast_tanh(cvv.z + cgr[2]) * whr[2];
                z += fast_tanh(cvv.w + cgr[3]) * whr[3];
            }
            z += fast_tanh(((const float*)cvp)[48] + cgl[tt][48]) * whl[48]; // a=48
            zbuf[tt][k] = z;
        }
    }
    __syncthreads();

    // --- softmax over K (each wave reduces 2 rows via wave32 shuffles) ---
    const int wave = tid >> 5;
    const int lane = tid & 31;
    for (int r = 0; r < 2; ++r) {
        const int row = wave * 2 + r;
        float vals[8];
        float m = -3.0e38f;
#pragma unroll
        for (int i = 0; i < 8; ++i) {
            vals[i] = zbuf[row][lane + (i << 5)];
            m = fmaxf(m, vals[i]);
        }
#pragma unroll
        for (int off = 16; off > 0; off >>= 1) m = fmaxf(m, __shfl_xor(m, off, 32));
        float s = 0.f;
#pragma unroll
        for (int i = 0; i < 8; ++i) { vals[i] = __expf(vals[i] - m); s += vals[i]; }
#pragma unroll
        for (int off = 16; off > 0; off >>= 1) s += __shfl_xor(s, off, 32);
        const float inv = 1.0f / s;
        float* arow = alpha + (size_t)(b * Tt + t0 + row) * Kk;
#pragma unroll
        for (int i = 0; i < 8; ++i) {
            const float av = vals[i] * inv;
            zbuf[row][lane + (i << 5)] = av;
            arow[lane + (i << 5)] = av;
        }
    }
    __syncthreads();

    // --- c_hat = alpha(16x256) * des_b(256x512), f32 WMMA ---
    const int lr = lane & 15;
    const int lh = lane >> 4;
    const float* desb = des + (size_t)b * Kk * Hh;
    for (int j = 0; j < 4; ++j) {                    // 8 waves * 4 = 32 N-tiles
        const int n0 = (wave * 4 + j) << 4;
        v8f c = {0.f, 0.f, 0.f, 0.f, 0.f, 0.f, 0.f, 0.f};
        for (int kb = 0; kb < Kk; kb += 4) {
            const int k = kb + (lh << 1);
            v2f a = *(const v2f*)&zbuf[lr][k];        // A-frag from LDS alpha
            const float* bp = desb + (size_t)k * Hh + n0 + lr;
            v2f bf; bf.x = bp[0]; bf.y = bp[Hh];      // B-frag: rows k, k+1
            c = __builtin_amdgcn_wmma_f32_16x16x4_f32(false, a, false, bf,
                                                      (short)0, c, false, false);
        }
        float* crow = chat + (size_t)(b * Tt + t0 + (lh << 3)) * Hh + n0 + lr;
#pragma unroll
        for (int v = 0; v < 8; ++v) crow[(size_t)v * Hh] = c[v];
    }
}

// ---------------------------------------------------------------------------
extern "C" void kernel_launch(void* const* d_in, const int* in_sizes, int n_in,
                              void* d_out, int out_size, void* d_ws, size_t ws_size,
                              hipStream_t stream) {
    const float* des = (const float*)d_in[0];   // (16,256,512)
    const float* tit = (const float*)d_in[1];   // (16,256,512)
    const float* Wv  = (const float*)d_in[2];   // (49,512)
    const float* Wg  = (const float*)d_in[3];   // (49,512)
    const float* Wh  = (const float*)d_in[4];   // (1,49)

    float* chat  = (float*)d_out;                           // (16,256,512)
    float* alpha = chat + (size_t)Bb * Tt * Hh;             // (16,256,256)

    float* cv = (float*)d_ws;                               // (16*256, 64)
    float* cg = cv + (size_t)Bb * Kk * AP;                  // (16*256, 64)

    // 1024 tiles / 8 waves per block = 128 blocks
    proj_wmma<<<128, 256, 0, stream>>>(des, Wv, cv);
    proj_wmma<<<128, 256, 0, stream>>>(tit, Wg, cg);

    // one block per (b, t-tile): 16 * 16 = 256 blocks
    attn_wmma<<<256, 256, 0, stream>>>(cv, cg, Wh, des, chat, alpha);
}